// SplitSpatial_87857851007288
// MI455X (gfx1250) — compile-verified
//
#include <hip/hip_runtime.h>

// Pixel-unshuffle: (B=16, C=64, H=256, W=256) f32 -> (B, 4C, H/2, W/2) f32
//   out group 0: rows even, cols even
//   out group 1: rows odd,  cols even
//   out group 2: rows even, cols odd
//   out group 3: rows odd,  cols odd
//
// Pure streaming permutation: 256 MiB in + 256 MiB out, 0 FLOPs.
// Strategy: per-thread 2x float4 NT loads from one input row (8 consecutive
// floats), deinterleave even/odd columns into two float4s, 2x float4 NT
// stores (each fully coalesced across the wave).

typedef float float4v __attribute__((ext_vector_type(4)));

namespace {
constexpr int kB  = 16;
constexpr int kC  = 64;
constexpr int kH  = 256;
constexpr int kW  = 256;
constexpr int kHo = kH / 2;   // 128
constexpr int kWo = kW / 2;   // 128
constexpr int kChunksPerRow = kW / 8;           // 32 threads per input row
constexpr long long kThreads =
    (long long)kB * kC * kH * kChunksPerRow;    // 8,388,608
}

__global__ __launch_bounds__(256)
void SplitSpatial_kernel(const float* __restrict__ in, float* __restrict__ out) {
  const unsigned t = blockIdx.x * blockDim.x + threadIdx.x;

  // Decode (b, c, r, wq) from flat id — all power-of-two shifts/masks.
  const unsigned wq = t & 31u;          // 8-float chunk within row: 0..31
  const unsigned r  = (t >> 5) & 255u;  // input row: 0..255
  const unsigned c  = (t >> 13) & 63u;  // channel: 0..63
  const unsigned b  = t >> 19;          // batch: 0..15

  // ---- Input: two B128 NT loads covering in[b,c,r, wq*8 .. wq*8+7] ----
  const size_t in_base =
      ((((size_t)b * kC + c) * kH + r) * kW) + (size_t)wq * 8;
  const float4v* __restrict__ ip =
      reinterpret_cast<const float4v*>(in + in_base);
  const float4v v0 = __builtin_nontemporal_load(ip);       // cols w..w+3
  const float4v v1 = __builtin_nontemporal_load(ip + 1);   // cols w+4..w+7

  // Deinterleave even/odd columns.
  const float4v even = {v0.x, v0.z, v1.x, v1.z};
  const float4v odd  = {v0.y, v0.w, v1.y, v1.w};

  // ---- Output placement ----
  const unsigned p     = r & 1u;        // row parity
  const unsigned h_out = r >> 1;        // 0..127
  const unsigned w_out = wq * 4;        // 0..124 step 4

  // group for even cols = p (0 or 1); group for odd cols = p + 2 (2 or 3)
  const size_t out_even =
      ((((size_t)b * (4 * kC) + (size_t)p * kC + c) * kHo + h_out) * kWo) + w_out;
  const size_t out_odd =
      ((((size_t)b * (4 * kC) + (size_t)(p + 2) * kC + c) * kHo + h_out) * kWo) + w_out;

  __builtin_nontemporal_store(even, reinterpret_cast<float4v*>(out + out_even));
  __builtin_nontemporal_store(odd,  reinterpret_cast<float4v*>(out + out_odd));
}

extern "C" void kernel_launch(void* const* d_in, const int* in_sizes, int n_in,
                              void* d_out, int out_size, void* d_ws, size_t ws_size,
                              hipStream_t stream) {
  (void)in_sizes; (void)n_in; (void)out_size; (void)d_ws; (void)ws_size;

  const float* x = reinterpret_cast<const float*>(d_in[0]);
  float* y = reinterpret_cast<float*>(d_out);

  constexpr int kBlock = 256;                       // 8 wave32s
  const int grid = (int)(kThreads / kBlock);        // 32768, exact cover
  SplitSpatial_kernel<<<grid, kBlock, 0, stream>>>(x, y);
}